// N2JNet_15513421873647
// MI455X (gfx1250) — compile-verified
//
#include <hip/hip_runtime.h>
#include <hip/hip_bf16.h>
#include <math.h>

typedef __attribute__((ext_vector_type(2))) float v2f;
typedef __attribute__((ext_vector_type(8))) float v8f;

#define NNODES 500000
#define NGRAPH 8192
#define NITER  20
#define TILES  5                 // 16-row tiles per wave; 80 nodes/block; 500000 % 80 == 0

__device__ inline v8f vzero8() { v8f z = {0.f,0.f,0.f,0.f,0.f,0.f,0.f,0.f}; return z; }

#if __has_builtin(__builtin_amdgcn_wmma_f32_16x16x4_f32)
__device__ inline v8f wmma4(v2f a, v2f b, v8f c) {
  // D(16x16,f32) = A(16x4,f32) x B(4x16,f32) + C
  return __builtin_amdgcn_wmma_f32_16x16x4_f32(false, a, false, b, (short)0, c, false, false);
}
#else
__device__ inline v8f wmma4(v2f a, v2f b, v8f c) { c[0] += a[0] * b[0]; return c; }
#endif

// ---------------------------------------------------------------------------
// Weight packing: B-operand layout for V_WMMA_F32_16X16X4_F32.
// lane l holds v0 = W[kt*4 + 2*(l>=16) + 0][nt*16 + (l&15)]
//              v1 = W[kt*4 + 2*(l>=16) + 1][nt*16 + (l&15)]
// flat: dst[kt*128 + nt*64 + lane*2 + j], zero-padded for cols >= 20.
// ---------------------------------------------------------------------------
__global__ void pack_w_kernel(const float* __restrict__ src, float* __restrict__ dst,
                              int K, int srcIterStride, int dstIterStride) {
  const int it  = blockIdx.y;
  const int idx = blockIdx.x * blockDim.x + threadIdx.x;
  const int total = (K / 4) * 128;
  if (idx >= total) return;
  const int j    = idx & 1;
  const int lane = (idx >> 1) & 31;
  const int nt   = (idx >> 6) & 1;
  const int kt   = idx >> 7;
  const int col  = nt * 16 + (lane & 15);
  const int row  = kt * 4 + 2 * (lane >> 4) + j;
  float v = (col < 20) ? src[(size_t)it * srcIterStride + row * 20 + col] : 0.0f;
  dst[(size_t)it * dstIterStride + idx] = v;
}

__global__ void zero_kernel(float* __restrict__ p, int n) {
  int i = blockIdx.x * blockDim.x + threadIdx.x;
  if (i < n) p[i] = 0.0f;
}

// ---------------------------------------------------------------------------
// WMMA building blocks
// ---------------------------------------------------------------------------
template <int KT>
__device__ inline void load_b(const float* __restrict__ pk, int lane, v2f* b0r, v2f* b1r) {
#pragma unroll
  for (int kt = 0; kt < KT; ++kt) {
    const float* bp = pk + (size_t)kt * 128 + lane * 2;
    b0r[kt].x = bp[0];  b0r[kt].y = bp[1];
    b1r[kt].x = bp[64]; b1r[kt].y = bp[65];
  }
}

template <int KT>
__device__ inline void gemm_tile_regB(const float* __restrict__ A, const v2f* b0r, const v2f* b1r,
                                      int ktBegin, int lane, v8f& c0, v8f& c1) {
  const int m  = lane & 15;
  const int hi = lane >> 4;
#pragma unroll
  for (int kt = 0; kt < KT; ++kt) {
    const float* ar = A + m * 21 + kt * 4 + 2 * hi;
    v2f a; a.x = ar[0]; a.y = ar[1];
    c0 = wmma4(a, b0r[ktBegin + kt], c0);
    c1 = wmma4(a, b1r[ktBegin + kt], c1);
  }
}

// C-tile (VGPR j holds rows j / j+8 for lane halves) + bias -> LDS H[16][33]
__device__ inline void store_bias(v8f c0, v8f c1, const float* __restrict__ bias,
                                  float* __restrict__ H, int lane) {
  const int colL = lane & 15;
  const int hi   = lane >> 4;
  const float bb0 = bias[colL];
  const float bb1 = (colL < 4) ? bias[16 + colL] : 0.0f;
#pragma unroll
  for (int j = 0; j < 8; ++j) {
    const int row = j + 8 * hi;
    H[row * 33 + colL]      = c0[j] + bb0;
    H[row * 33 + 16 + colL] = c1[j] + bb1;
  }
}

// LayerNorm row statistics for all TILES*16 rows
__device__ inline void all_stats(const float* __restrict__ Hb, float* __restrict__ statM,
                                 float* __restrict__ statR, int lane) {
  for (int q = 0; q < (TILES * 16 + 31) / 32; ++q) {
    int r = lane + 32 * q;
    if (r < TILES * 16) {
      const float* Hrow = Hb + (r / 16) * 528 + (r % 16) * 33;
      float s = 0.f, s2 = 0.f;
#pragma unroll
      for (int c = 0; c < 20; ++c) { float h = Hrow[c]; s += h; s2 += h * h; }
      float m = s * 0.05f;
      float v = s2 * 0.05f - m * m;
      statM[r] = m;
      statR[r] = rsqrtf(v + 1e-5f);
    }
  }
}

struct NPParams {
  const float *pkNW0, *pkNW1, *pkNW2;   // node MLP packed weights (this iteration)
  const float *pkPW0, *pkPW1, *pkPW2;   // pre  MLP packed weights (this iteration)
  const float *nb0, *ng0, *nbe0, *nb1, *ng1, *nbe1, *nb2, *ng2, *nbe2;
  const float *pb0, *pb1, *pb2, *pg, *pbe;
};

// One wave per block; TILES 16-node tiles per wave, layer-major with B in registers.
__global__ __launch_bounds__(32) void node_pre_kernel(
    float* __restrict__ xl, const float* __restrict__ u, const int* __restrict__ batch,
    float* __restrict__ agg, NPParams P) {
  __shared__ float Xb[TILES * 336];   // [tile][16][21] current x
  __shared__ float Ub[TILES * 336];   // [tile][16][21] u[batch]
  __shared__ float Cb[TILES * 336];   // [tile][16][21] activations
  __shared__ float Hb[TILES * 528];   // [tile][16][33] raw linear outputs
  __shared__ float statM[TILES * 16];
  __shared__ float statR[TILES * 16];
  __shared__ int   gI[TILES * 16];

  const int lane = threadIdx.x;
  const int n0   = blockIdx.x * (16 * TILES);

#pragma unroll
  for (int tt = 0; tt < TILES; ++tt)
    if (lane < 16) gI[tt * 16 + lane] = batch[n0 + tt * 16 + lane];
#pragma unroll
  for (int t = 0; t < 10 * TILES; ++t) {   // 1600 = 80 rows x 20 cols, coalesced
    int i = lane + 32 * t;
    int tt = i / 320, li = i % 320, row = li / 20, col = li % 20;
    Xb[tt * 336 + row * 21 + col] = xl[n0 * 20 + i];
  }
  __syncthreads();
#pragma unroll
  for (int t = 0; t < 10 * TILES; ++t) {
    int i = lane + 32 * t;
    int tt = i / 320, li = i % 320, row = li / 20, col = li % 20;
    Ub[tt * 336 + row * 21 + col] = u[gI[tt * 16 + row] * 20 + col];
  }
  __syncthreads();

  v2f b0r[10], b1r[10];

  // ================= node MLP: (Lin->LN->ReLU) x2, Lin->LN, +residual =======
  // Layer N1: cat(x,u) 40 -> 20
  load_b<10>(P.pkNW0, lane, b0r, b1r);
#pragma unroll
  for (int tt = 0; tt < TILES; ++tt) {
    v8f c0 = vzero8(), c1 = vzero8();
    gemm_tile_regB<5>(Xb + tt * 336, b0r, b1r, 0, lane, c0, c1);
    gemm_tile_regB<5>(Ub + tt * 336, b0r, b1r, 5, lane, c0, c1);
    store_bias(c0, c1, P.nb0, Hb + tt * 528, lane);
  }
  __syncthreads();
  all_stats(Hb, statM, statR, lane);
  __syncthreads();
#pragma unroll
  for (int t = 0; t < 10 * TILES; ++t) {
    int i = lane + 32 * t;
    int tt = i / 320, li = i % 320, row = li / 20, col = li % 20, r = tt * 16 + row;
    float o = (Hb[tt * 528 + row * 33 + col] - statM[r]) * statR[r] * P.ng0[col] + P.nbe0[col];
    Cb[tt * 336 + row * 21 + col] = fmaxf(o, 0.f);
  }
  __syncthreads();

  // Layer N2: 20 -> 20
  load_b<5>(P.pkNW1, lane, b0r, b1r);
#pragma unroll
  for (int tt = 0; tt < TILES; ++tt) {
    v8f c0 = vzero8(), c1 = vzero8();
    gemm_tile_regB<5>(Cb + tt * 336, b0r, b1r, 0, lane, c0, c1);
    store_bias(c0, c1, P.nb1, Hb + tt * 528, lane);
  }
  __syncthreads();
  all_stats(Hb, statM, statR, lane);
  __syncthreads();
#pragma unroll
  for (int t = 0; t < 10 * TILES; ++t) {
    int i = lane + 32 * t;
    int tt = i / 320, li = i % 320, row = li / 20, col = li % 20, r = tt * 16 + row;
    float o = (Hb[tt * 528 + row * 33 + col] - statM[r]) * statR[r] * P.ng1[col] + P.nbe1[col];
    Cb[tt * 336 + row * 21 + col] = fmaxf(o, 0.f);
  }
  __syncthreads();

  // Layer N3: 20 -> 20, LN, residual into Xb and global xl
  load_b<5>(P.pkNW2, lane, b0r, b1r);
#pragma unroll
  for (int tt = 0; tt < TILES; ++tt) {
    v8f c0 = vzero8(), c1 = vzero8();
    gemm_tile_regB<5>(Cb + tt * 336, b0r, b1r, 0, lane, c0, c1);
    store_bias(c0, c1, P.nb2, Hb + tt * 528, lane);
  }
  __syncthreads();
  all_stats(Hb, statM, statR, lane);
  __syncthreads();
#pragma unroll
  for (int t = 0; t < 10 * TILES; ++t) {
    int i = lane + 32 * t;
    int tt = i / 320, li = i % 320, row = li / 20, col = li % 20, r = tt * 16 + row;
    float o  = (Hb[tt * 528 + row * 33 + col] - statM[r]) * statR[r] * P.ng2[col] + P.nbe2[col];
    float xn = Xb[tt * 336 + row * 21 + col] + o;
    Xb[tt * 336 + row * 21 + col] = xn;
    xl[n0 * 20 + i] = xn;
  }
  __syncthreads();

  // ================= pre MLP: Lin->ReLU x2, Lin->LN; scatter-add ============
  // Layer P1: cat(x_new, u) 40 -> 20
  load_b<10>(P.pkPW0, lane, b0r, b1r);
#pragma unroll
  for (int tt = 0; tt < TILES; ++tt) {
    v8f c0 = vzero8(), c1 = vzero8();
    gemm_tile_regB<5>(Xb + tt * 336, b0r, b1r, 0, lane, c0, c1);
    gemm_tile_regB<5>(Ub + tt * 336, b0r, b1r, 5, lane, c0, c1);
    store_bias(c0, c1, P.pb0, Hb + tt * 528, lane);
  }
  __syncthreads();
#pragma unroll
  for (int t = 0; t < 10 * TILES; ++t) {
    int i = lane + 32 * t;
    int tt = i / 320, li = i % 320, row = li / 20, col = li % 20;
    Cb[tt * 336 + row * 21 + col] = fmaxf(Hb[tt * 528 + row * 33 + col], 0.f);
  }
  __syncthreads();

  // Layer P2: 20 -> 20
  load_b<5>(P.pkPW1, lane, b0r, b1r);
#pragma unroll
  for (int tt = 0; tt < TILES; ++tt) {
    v8f c0 = vzero8(), c1 = vzero8();
    gemm_tile_regB<5>(Cb + tt * 336, b0r, b1r, 0, lane, c0, c1);
    store_bias(c0, c1, P.pb1, Hb + tt * 528, lane);
  }
  __syncthreads();
#pragma unroll
  for (int t = 0; t < 10 * TILES; ++t) {
    int i = lane + 32 * t;
    int tt = i / 320, li = i % 320, row = li / 20, col = li % 20;
    Cb[tt * 336 + row * 21 + col] = fmaxf(Hb[tt * 528 + row * 33 + col], 0.f);
  }
  __syncthreads();

  // Layer P3: 20 -> 20, LN, atomic scatter-add into per-graph agg
  load_b<5>(P.pkPW2, lane, b0r, b1r);
#pragma unroll
  for (int tt = 0; tt < TILES; ++tt) {
    v8f c0 = vzero8(), c1 = vzero8();
    gemm_tile_regB<5>(Cb + tt * 336, b0r, b1r, 0, lane, c0, c1);
    store_bias(c0, c1, P.pb2, Hb + tt * 528, lane);
  }
  __syncthreads();
  all_stats(Hb, statM, statR, lane);
  __syncthreads();
#pragma unroll
  for (int t = 0; t < 10 * TILES; ++t) {
    int i = lane + 32 * t;
    int tt = i / 320, li = i % 320, row = li / 20, col = li % 20, r = tt * 16 + row;
    float o = (Hb[tt * 528 + row * 33 + col] - statM[r]) * statR[r] * P.pg[col] + P.pbe[col];
    atomicAdd(&agg[gI[r] * 20 + col], o);
  }
}

// ---------------------------------------------------------------------------
// Scalar kernels for the cheap parts (<3% of FLOPs)
// ---------------------------------------------------------------------------
struct MlpP { const float *W0, *b0, *W1, *b1, *W2, *b2, *g, *be; };
struct OutP { const float *W0, *b0, *W1, *b1, *W2, *b2; };

template <int DIN>
__global__ void init_mlp_kernel(const float* __restrict__ xin, MlpP p,
                                float* __restrict__ out, int n) {
  int i = blockIdx.x * blockDim.x + threadIdx.x;
  if (i >= n) return;
  float in[DIN];
#pragma unroll
  for (int c = 0; c < DIN; ++c) in[c] = xin[(size_t)i * DIN + c];
  float h0[20], h1[20], h2[20];
#pragma unroll
  for (int o = 0; o < 20; ++o) {
    float s = p.b0[o];
#pragma unroll
    for (int k = 0; k < DIN; ++k) s += in[k] * p.W0[k * 20 + o];
    h0[o] = fmaxf(s, 0.f);
  }
#pragma unroll
  for (int o = 0; o < 20; ++o) {
    float s = p.b1[o];
#pragma unroll
    for (int k = 0; k < 20; ++k) s += h0[k] * p.W1[k * 20 + o];
    h1[o] = fmaxf(s, 0.f);
  }
#pragma unroll
  for (int o = 0; o < 20; ++o) {
    float s = p.b2[o];
#pragma unroll
    for (int k = 0; k < 20; ++k) s += h1[k] * p.W2[k * 20 + o];
    h2[o] = s;
  }
  float m = 0.f;
#pragma unroll
  for (int o = 0; o < 20; ++o) m += h2[o];
  m *= 0.05f;
  float v = 0.f;
#pragma unroll
  for (int o = 0; o < 20; ++o) { float d = h2[o] - m; v += d * d; }
  float r = rsqrtf(v * 0.05f + 1e-5f);
#pragma unroll
  for (int o = 0; o < 20; ++o) out[(size_t)i * 20 + o] = (h2[o] - m) * r * p.g[o] + p.be[o];
}

__global__ void post_mlp_kernel(const float* __restrict__ agg, float* __restrict__ u,
                                MlpP p, int n) {
  int i = blockIdx.x * blockDim.x + threadIdx.x;
  if (i >= n) return;
  float in[40];
#pragma unroll
  for (int c = 0; c < 20; ++c) in[c] = agg[i * 20 + c];
#pragma unroll
  for (int c = 0; c < 20; ++c) in[20 + c] = u[i * 20 + c];
  float h0[20], h1[20], h2[20];
#pragma unroll
  for (int o = 0; o < 20; ++o) {
    float s = p.b0[o];
#pragma unroll
    for (int k = 0; k < 40; ++k) s += in[k] * p.W0[k * 20 + o];
    h0[o] = fmaxf(s, 0.f);
  }
#pragma unroll
  for (int o = 0; o < 20; ++o) {
    float s = p.b1[o];
#pragma unroll
    for (int k = 0; k < 20; ++k) s += h0[k] * p.W1[k * 20 + o];
    h1[o] = fmaxf(s, 0.f);
  }
#pragma unroll
  for (int o = 0; o < 20; ++o) {
    float s = p.b2[o];
#pragma unroll
    for (int k = 0; k < 20; ++k) s += h1[k] * p.W2[k * 20 + o];
    h2[o] = s;
  }
  float m = 0.f;
#pragma unroll
  for (int o = 0; o < 20; ++o) m += h2[o];
  m *= 0.05f;
  float v = 0.f;
#pragma unroll
  for (int o = 0; o < 20; ++o) { float d = h2[o] - m; v += d * d; }
  float r = rsqrtf(v * 0.05f + 1e-5f);
#pragma unroll
  for (int o = 0; o < 20; ++o) u[i * 20 + o] += (h2[o] - m) * r * p.g[o] + p.be[o];
}

__global__ void out_mlp_kernel(const float* __restrict__ xin, OutP p,
                               float* __restrict__ out, int n) {
  int i = blockIdx.x * blockDim.x + threadIdx.x;
  if (i >= n) return;
  float in[20], h0[20], h1[20];
#pragma unroll
  for (int c = 0; c < 20; ++c) in[c] = xin[(size_t)i * 20 + c];
#pragma unroll
  for (int o = 0; o < 20; ++o) {
    float s = p.b0[o];
#pragma unroll
    for (int k = 0; k < 20; ++k) s += in[k] * p.W0[k * 20 + o];
    h0[o] = fmaxf(s, 0.f);
  }
#pragma unroll
  for (int o = 0; o < 20; ++o) {
    float s = p.b1[o];
#pragma unroll
    for (int k = 0; k < 20; ++k) s += h0[k] * p.W1[k * 20 + o];
    h1[o] = fmaxf(s, 0.f);
  }
#pragma unroll
  for (int o = 0; o < 2; ++o) {
    float s = p.b2[o];
#pragma unroll
    for (int k = 0; k < 20; ++k) s += h1[k] * p.W2[k * 2 + o];
    out[(size_t)i * 2 + o] = s;
  }
}

// ---------------------------------------------------------------------------
// Host launcher
// ---------------------------------------------------------------------------
extern "C" void kernel_launch(void* const* d_in, const int* in_sizes, int n_in,
                              void* d_out, int out_size, void* d_ws, size_t ws_size,
                              hipStream_t stream) {
  auto F = [&](int i) { return reinterpret_cast<const float*>(d_in[i]); };
  const float* x      = F(0);                                  // [N,8]
  const float* x_meta = F(1);                                  // [B,2]
  const int*   batch  = reinterpret_cast<const int*>(d_in[2]); // [N]

  // Input order: insertion order of setup_inputs() params dict.
  // 3..10 node_init, 11..18 global_init, 19..30 node, 31..38 pre,
  // 39..46 post, 47..52 out_local, 53..58 out_global.

  float* ws    = reinterpret_cast<float*>(d_ws);
  float* xl    = ws;                                  // [N,20]
  float* u     = xl + (size_t)NNODES * 20;            // [B,20]
  float* agg   = u + (size_t)NGRAPH * 20;             // [B,20]
  float* pkNW0 = agg + (size_t)NGRAPH * 20;           // 20*1280
  float* pkNW1 = pkNW0 + NITER * 1280;                // 20*640
  float* pkNW2 = pkNW1 + NITER * 640;
  float* pkPW0 = pkNW2 + NITER * 640;                 // 20*1280
  float* pkPW1 = pkPW0 + NITER * 1280;
  float* pkPW2 = pkPW1 + NITER * 640;

  // Pack weights into WMMA B-operand layout (once per launch).
  pack_w_kernel<<<dim3(5, NITER), 256, 0, stream>>>(F(19), pkNW0, 40, 800, 1280);
  pack_w_kernel<<<dim3(3, NITER), 256, 0, stream>>>(F(23), pkNW1, 20, 400, 640);
  pack_w_kernel<<<dim3(3, NITER), 256, 0, stream>>>(F(27), pkNW2, 20, 400, 640);
  pack_w_kernel<<<dim3(5, NITER), 256, 0, stream>>>(F(31), pkPW0, 40, 800, 1280);
  pack_w_kernel<<<dim3(3, NITER), 256, 0, stream>>>(F(33), pkPW1, 20, 400, 640);
  pack_w_kernel<<<dim3(3, NITER), 256, 0, stream>>>(F(35), pkPW2, 20, 400, 640);

  // Init MLPs.
  MlpP niP{F(3), F(4), F(5), F(6), F(7), F(8), F(9), F(10)};
  init_mlp_kernel<8><<<(NNODES + 255) / 256, 256, 0, stream>>>(x, niP, xl, NNODES);
  MlpP giP{F(11), F(12), F(13), F(14), F(15), F(16), F(17), F(18)};
  init_mlp_kernel<2><<<(NGRAPH + 255) / 256, 256, 0, stream>>>(x_meta, giP, u, NGRAPH);

  // Message-passing iterations.
  for (int it = 0; it < NITER; ++it) {
    zero_kernel<<<(NGRAPH * 20 + 255) / 256, 256, 0, stream>>>(agg, NGRAPH * 20);

    NPParams np;
    np.pkNW0 = pkNW0 + (size_t)it * 1280;
    np.pkNW1 = pkNW1 + (size_t)it * 640;
    np.pkNW2 = pkNW2 + (size_t)it * 640;
    np.pkPW0 = pkPW0 + (size_t)it * 1280;
    np.pkPW1 = pkPW1 + (size_t)it * 640;
    np.pkPW2 = pkPW2 + (size_t)it * 640;
    np.nb0 = F(20) + it * 20; np.ng0 = F(21) + it * 20; np.nbe0 = F(22) + it * 20;
    np.nb1 = F(24) + it * 20; np.ng1 = F(25) + it * 20; np.nbe1 = F(26) + it * 20;
    np.nb2 = F(28) + it * 20; np.ng2 = F(29) + it * 20; np.nbe2 = F(30) + it * 20;
    np.pb0 = F(32) + it * 20; np.pb1 = F(34) + it * 20; np.pb2 = F(36) + it * 20;
    np.pg  = F(37) + it * 20; np.pbe = F(38) + it * 20;

    node_pre_kernel<<<NNODES / (16 * TILES), 32, 0, stream>>>(xl, u, batch, agg, np);

    MlpP pp{F(39) + (size_t)it * 800, F(40) + it * 20, F(41) + (size_t)it * 400,
            F(42) + it * 20, F(43) + (size_t)it * 400, F(44) + it * 20,
            F(45) + it * 20, F(46) + it * 20};
    post_mlp_kernel<<<(NGRAPH + 255) / 256, 256, 0, stream>>>(agg, u, pp, NGRAPH);
  }

  // Output heads.
  float* out = reinterpret_cast<float*>(d_out);
  OutP olP{F(47), F(48), F(49), F(50), F(51), F(52)};
  out_mlp_kernel<<<(NNODES + 255) / 256, 256, 0, stream>>>(xl, olP, out, NNODES);
  OutP ogP{F(53), F(54), F(55), F(56), F(57), F(58)};
  out_mlp_kernel<<<(NGRAPH + 255) / 256, 256, 0, stream>>>(u, ogP, out + (size_t)NNODES * 2, NGRAPH);
}